// DiagLaplacianBuilder_18459769438524
// MI455X (gfx1250) — compile-verified
//
#include <hip/hip_runtime.h>
#include <hip/hip_bf16.h>
#include <cstdint>
#include <cstddef>

#define DD 4          // maps feature dim (reference constant)
#define SCAN_T 1024

typedef __attribute__((address_space(3))) float4 lds_f4_t;

// ---------------- zero scratch ----------------
__global__ void zero_i32_kernel(int* __restrict__ p, int n) {
  int i = blockIdx.x * blockDim.x + threadIdx.x;
  int stride = gridDim.x * blockDim.x;
  for (; i < n; i += stride) p[i] = 0;
}

// ---------------- node degrees ----------------
__global__ void degree_kernel(const int* __restrict__ edge_row, int* __restrict__ deg, int twoE) {
  int i = blockIdx.x * blockDim.x + threadIdx.x;
  if (i < twoE) atomicAdd(&deg[edge_row[i]], 1);
}

// -------- one-block exclusive scan: A[n]=prefix(deg), B[n]=DD*prefix(deg+1) --------
__global__ __launch_bounds__(SCAN_T) void scan_kernel(const int* __restrict__ deg,
                                                      int* __restrict__ A,
                                                      int* __restrict__ B, int N) {
  __shared__ int pA[SCAN_T];
  __shared__ int pB[SCAN_T];
  int t = threadIdx.x;
  int chunk = (N + SCAN_T - 1) / SCAN_T;
  int lo = t * chunk;
  int hi = min(N, lo + chunk);
  int sA = 0, sB = 0;
  for (int i = lo; i < hi; ++i) { int d = deg[i]; sA += d; sB += d + 1; }
  pA[t] = sA; pB[t] = sB;
  __syncthreads();
  if (t == 0) {
    int aA = 0, aB = 0;
    for (int i = 0; i < SCAN_T; ++i) {
      int ta = pA[i], tb = pB[i];
      pA[i] = aA; pB[i] = aB;
      aA += ta; aB += tb;
    }
  }
  __syncthreads();
  int bA = pA[t], bB = pB[t];
  for (int i = lo; i < hi; ++i) {
    A[i] = bA;
    B[i] = DD * bB;
    int d = deg[i];
    bA += d; bB += d + 1;
  }
}

// -------- fill directed adjacency (CSR buckets; slot order racy but consumers are order-free) ----
__global__ void adj_fill_kernel(const int* __restrict__ edge_row, const int* __restrict__ A,
                                int* __restrict__ cursor, int* __restrict__ adj_owner,
                                int* __restrict__ adj_nbr, int* __restrict__ adj_edge, int E) {
  int e = blockIdx.x * blockDim.x + threadIdx.x;
  if (e >= E) return;
  int lo = edge_row[e];
  int hi = edge_row[e + E];
  int s0 = atomicAdd(&cursor[lo], 1);
  int i0 = A[lo] + s0;
  adj_owner[i0] = lo; adj_nbr[i0] = hi; adj_edge[i0] = e;
  int s1 = atomicAdd(&cursor[hi], 1);
  int i1 = A[hi] + s1;
  adj_owner[i1] = hi; adj_nbr[i1] = lo; adj_edge[i1] = e;
}

// -------- off-diagonal scatter: rank by counting (order-independent) → CSR position ----------
__global__ void scatter_offdiag_kernel(const float* __restrict__ maps,
                                       const int* __restrict__ left_idx,
                                       const int* __restrict__ right_idx,
                                       const int* __restrict__ deg, const int* __restrict__ A,
                                       const int* __restrict__ B,
                                       const int* __restrict__ adj_owner,
                                       const int* __restrict__ adj_nbr,
                                       const int* __restrict__ adj_edge,
                                       float* __restrict__ rows, float* __restrict__ cols,
                                       float* __restrict__ vals, int twoE) {
  int i = blockIdx.x * blockDim.x + threadIdx.x;
  if (i >= twoE) return;
  int n = adj_owner[i];
  int m = adj_nbr[i];
  int e = adj_edge[i];
  int a0 = A[n];
  int dg = deg[n];
  __builtin_prefetch(adj_nbr + a0, 0, 1);        // global_prefetch_b8: pull segment before scan
  int rank = 0;
  for (int j = a0; j < a0 + dg; ++j) rank += (adj_nbr[j] < m) ? 1 : 0;
  int pos = rank + ((m > n) ? 1 : 0);   // diagonal sits where a neighbor id == n would
  const float4 l = *(const float4*)(maps + (size_t)left_idx[e] * DD);
  const float4 r = *(const float4*)(maps + (size_t)right_idx[e] * DD);
  float v[DD] = { -(l.x * r.x), -(l.y * r.y), -(l.z * r.z), -(l.w * r.w) };
  int base = B[n];
  int rl = dg + 1;
#pragma unroll
  for (int k = 0; k < DD; ++k) {
    int o = base + k * rl + pos;
    rows[o] = (float)(n * DD + k);
    cols[o] = (float)(m * DD + k);
    vals[o] = v[k];
  }
}

// -------- diagonal: deterministic per-node sum in sorted-neighbor order (no float atomics) ------
__global__ void diag_kernel(const float* __restrict__ maps, const int* __restrict__ deg,
                            const int* __restrict__ A, const int* __restrict__ B,
                            const int* __restrict__ adj_nbr, const int* __restrict__ adj_edge,
                            float* __restrict__ rows, float* __restrict__ cols,
                            float* __restrict__ vals, int N, int E) {
  int n = blockIdx.x * blockDim.x + threadIdx.x;
  if (n >= N) return;
  int dg = deg[n];
  int a0 = A[n];
  __builtin_prefetch(adj_nbr + a0, 0, 1);        // global_prefetch_b8 ahead of O(deg^2) selection
  float acc0 = 0.f, acc1 = 0.f, acc2 = 0.f, acc3 = 0.f;
  int cl = 0;      // count of neighbors with id < n  → diagonal slot
  int prev = -1;
  for (int r = 0; r < dg; ++r) {           // selection: visit neighbors in ascending id
    int best = 0x7fffffff, bj = a0;
    for (int j = a0; j < a0 + dg; ++j) {
      int mm = adj_nbr[j];
      if (mm > prev && mm < best) { best = mm; bj = j; }
    }
    prev = best;
    if (best < n) ++cl;
    int e = adj_edge[bj];
    int mrow = (best > n) ? e : (E + e);   // segment_sum contribution row in maps
    const float4 mv = *(const float4*)(maps + (size_t)mrow * DD);
    acc0 += mv.x * mv.x; acc1 += mv.y * mv.y; acc2 += mv.z * mv.z; acc3 += mv.w * mv.w;
  }
  float accs[DD] = { acc0, acc1, acc2, acc3 };
  int base = B[n];
  int rl = dg + 1;
#pragma unroll
  for (int k = 0; k < DD; ++k) {
    int o = base + k * rl + cl;
    float rc = (float)(n * DD + k);
    rows[o] = rc;
    cols[o] = rc;
    vals[o] = accs[k];
  }
}

// -------- saved_tril_maps: full async DMA round-trip (HBM→LDS→HBM, ASYNCcnt path) ---------------
__global__ __launch_bounds__(256) void tril_saved_kernel(const float* __restrict__ maps,
                                                         const int* __restrict__ left_idx,
                                                         const int* __restrict__ right_idx,
                                                         float* __restrict__ out, int E) {
  __shared__ float4 shL[256];
  __shared__ float4 shR[256];
  __shared__ float4 shV[256];
  int t = threadIdx.x;
  int e = blockIdx.x * 256 + t;
  int ec = min(e, E - 1);                  // keep all lanes valid for the async issue
  unsigned loff = (unsigned)left_idx[ec]  * (unsigned)(DD * sizeof(float));
  unsigned roff = (unsigned)right_idx[ec] * (unsigned)(DD * sizeof(float));
  lds_f4_t* lp = (lds_f4_t*)&shL[t];
  lds_f4_t* rp = (lds_f4_t*)&shR[t];
  // GVS mode: SGPR64 base + per-lane 32-bit byte offset → LDS, tracked with ASYNCcnt.
  asm volatile("global_load_async_to_lds_b128 %0, %1, %2"
               :: "v"(lp), "v"(loff), "s"(maps) : "memory");
  asm volatile("global_load_async_to_lds_b128 %0, %1, %2"
               :: "v"(rp), "v"(roff), "s"(maps) : "memory");
  asm volatile("s_wait_asynccnt 0" ::: "memory");   // wave-private tiles: no barrier needed
  float4 l = shL[t];
  float4 r = shR[t];
  float4 v;
  v.x = -(l.x * r.x); v.y = -(l.y * r.y); v.z = -(l.z * r.z); v.w = -(l.w * r.w);
  shV[t] = v;                                        // ds_store_b128
  // DS and ASYNC are separate counters: order the LDS write before the async DMA reads it.
  asm volatile("s_wait_dscnt 0x0" ::: "memory");
  if (e < E) {
    unsigned goff = (unsigned)e * (unsigned)(DD * sizeof(float));
    lds_f4_t* vp = (lds_f4_t*)&shV[t];
    // LDS → HBM async store (EXEC-masked); s_endpgm's implicit wait-idle drains it.
    asm volatile("global_store_async_from_lds_b128 %0, %1, %2"
                 :: "v"(goff), "v"(vp), "s"(out) : "memory");
  }
}

extern "C" void kernel_launch(void* const* d_in, const int* in_sizes, int n_in,
                              void* d_out, int out_size, void* d_ws, size_t ws_size,
                              hipStream_t stream) {
  const float* maps     = (const float*)d_in[0];
  const int* edge_row   = (const int*)d_in[1];
  const int* left_idx   = (const int*)d_in[2];
  const int* right_idx  = (const int*)d_in[3];
  // d_in[4] (tril_indices) / d_in[5] (diag_indices) structure is derived from edge_row.
  const int twoE = in_sizes[1];
  const int E    = in_sizes[2];
  const int N    = in_sizes[5] / (2 * DD);   // diag_indices has 2*N*D entries
  const int nnz  = twoE * DD + N * DD;

  // workspace carve-up (all int32)
  char* ws = (char*)d_ws;
  int* deg       = (int*)ws;  ws += (size_t)N * sizeof(int);
  int* A         = (int*)ws;  ws += (size_t)N * sizeof(int);
  int* B         = (int*)ws;  ws += (size_t)N * sizeof(int);
  int* cursor    = (int*)ws;  ws += (size_t)N * sizeof(int);
  int* adj_owner = (int*)ws;  ws += (size_t)twoE * sizeof(int);
  int* adj_nbr   = (int*)ws;  ws += (size_t)twoE * sizeof(int);
  int* adj_edge  = (int*)ws;  ws += (size_t)twoE * sizeof(int);
  (void)ws_size; (void)n_in; (void)out_size;

  float* rows  = (float*)d_out;
  float* cols  = rows + (size_t)nnz;
  float* vals  = cols + (size_t)nnz;
  float* saved = vals + (size_t)nnz;

  zero_i32_kernel<<<256, 256, 0, stream>>>(deg, N);
  zero_i32_kernel<<<256, 256, 0, stream>>>(cursor, N);

  degree_kernel<<<(twoE + 255) / 256, 256, 0, stream>>>(edge_row, deg, twoE);

  scan_kernel<<<1, SCAN_T, 0, stream>>>(deg, A, B, N);

  adj_fill_kernel<<<(E + 255) / 256, 256, 0, stream>>>(edge_row, A, cursor,
                                                       adj_owner, adj_nbr, adj_edge, E);

  scatter_offdiag_kernel<<<(twoE + 255) / 256, 256, 0, stream>>>(
      maps, left_idx, right_idx, deg, A, B, adj_owner, adj_nbr, adj_edge,
      rows, cols, vals, twoE);

  diag_kernel<<<(N + 255) / 256, 256, 0, stream>>>(maps, deg, A, B, adj_nbr, adj_edge,
                                                   rows, cols, vals, N, E);

  tril_saved_kernel<<<(E + 255) / 256, 256, 0, stream>>>(maps, left_idx, right_idx, saved, E);
}